// HEXLoss_66640712564868
// MI455X (gfx1250) — compile-verified
//
#include <hip/hip_runtime.h>

// HEX loss on MI455X (gfx1250, wave32).
//
// Math: the reference chain-CRF potential factorizes (pot[k,i,j] =
// exp(i*a_k)*exp(j*b_k)), so P(x_v=1) = sigmoid(fs[b,v]) and
//   loss = mean_b softplus(-fs[b, labels[b]]).
// This is a latency-bound gather + reduction (~2MB of cacheline traffic at
// 23.3 TB/s). We do the per-wave reduction on the matrix unit with
// V_WMMA_F32_16X16X4_F32 (fp32 in, fp32 accumulate -> zero precision loss):
// A = 64 softplus values (2 per lane, exactly the 16x4 fp32 A footprint),
// B = ones, so D[m,n] = row-sum of A. Wave total = (sum of 8 C regs in a
// lane<16) + (same in lane+16), combined through LDS.

typedef __attribute__((ext_vector_type(2))) float v2f;
typedef __attribute__((ext_vector_type(8))) float v8f;

#define HEX_B 32768
#define HEX_V 256
#define BLK_THREADS 256
#define ELEMS_PER_THREAD 2
#define NUM_BLOCKS (HEX_B / (BLK_THREADS * ELEMS_PER_THREAD))   // 64

__device__ __forceinline__ float softplus_neg(float x) {
  // softplus(-x) = max(-x, 0) + log1p(exp(-|x|)), numerically stable
  return fmaxf(-x, 0.0f) + log1pf(expf(-fabsf(x)));
}

__global__ void hex_partial_kernel(const float* __restrict__ fs,
                                   const int* __restrict__ labels,
                                   float* __restrict__ partials) {
  __shared__ float red[BLK_THREADS];
  const int tid = threadIdx.x;
  const int gid = blockIdx.x * BLK_THREADS + tid;

  // Each thread gathers 2 batch elements and computes softplus(-x) in fp32.
  float v0, v1;
  {
    const int b0 = gid * 2;
    const int b1 = b0 + 1;
    const float x0 = fs[b0 * HEX_V + labels[b0]];
    const float x1 = fs[b1 * HEX_V + labels[b1]];
    v0 = softplus_neg(x0);
    v1 = softplus_neg(x1);
  }

  // Wave-level reduction on the matrix unit: D = A x ones + C, fp32 end to end.
  v2f a;    a.x = v0;   a.y = v1;
  v2f ones; ones.x = 1.0f; ones.y = 1.0f;
  v8f c = {};
  // 8 args: (neg_a, A, neg_b, B, c_mod, C, reuse_a, reuse_b)
  c = __builtin_amdgcn_wmma_f32_16x16x4_f32(
      /*neg_a=*/false, a, /*neg_b=*/false, ones,
      /*c_mod=*/(short)0, c, /*reuse_a=*/false, /*reuse_b=*/false);

  // Every column of D is identical (B = ones). Lane n<16 carries D[0..7,n],
  // lane n+16 carries D[8..15,n]; sum of the 8 components in one low lane
  // plus one high lane is the full 64-value wave sum.
  float s = c[0] + c[1] + c[2] + c[3] + c[4] + c[5] + c[6] + c[7];
  red[tid] = s;
  __syncthreads();

  if (tid == 0) {
    float t = 0.0f;
    // 8 waves per block of 256 threads (wave32): combine lane0 + lane16 of each.
    #pragma unroll
    for (int w = 0; w < BLK_THREADS; w += 32) t += red[w] + red[w + 16];
    partials[blockIdx.x] = t;
  }
}

__global__ void hex_final_kernel(const float* __restrict__ partials,
                                 float* __restrict__ out) {
  if (threadIdx.x == 0) {
    float t = 0.0f;
    #pragma unroll
    for (int i = 0; i < NUM_BLOCKS; ++i) t += partials[i];
    out[0] = t * (1.0f / (float)HEX_B);
  }
}

extern "C" void kernel_launch(void* const* d_in, const int* in_sizes, int n_in,
                              void* d_out, int out_size, void* d_ws, size_t ws_size,
                              hipStream_t stream) {
  const float* fs     = (const float*)d_in[0];   // (B, V) float32
  const int*   labels = (const int*)d_in[1];     // (B,) integer
  float* out      = (float*)d_out;               // scalar loss
  float* partials = (float*)d_ws;                // NUM_BLOCKS fp32 partials

  hex_partial_kernel<<<NUM_BLOCKS, BLK_THREADS, 0, stream>>>(fs, labels, partials);
  hex_final_kernel<<<1, 32, 0, stream>>>(partials, out);
}